// GatedGraphConv_26216480375295
// MI455X (gfx1250) — compile-verified
//
#include <hip/hip_runtime.h>
#include <hip/hip_bf16.h>

// ---------------------------------------------------------------------------
// GatedGraphConv on MI455X (gfx1250, wave32).
//   h=feat; 5x { m[e]=h[src[e]] @ W[etype[e]]; a=segsum(m,dst); h=GRU(a,h) }
// Edge path: grid-stride wave-per-edge, LDS etype weight tables, f32 atomics.
// GRU path : V_WMMA_F32_16X16X4_F32 chains with fragment-major LDS weights
//            (each B operand = one conflict-free ds_load_b64 into an aligned
//             VGPR pair -> minimal mov churn around the WMMA chain).
// ---------------------------------------------------------------------------

typedef __attribute__((ext_vector_type(2))) float v2f;
typedef __attribute__((ext_vector_type(8))) float v8f;

#define OUTF 32
#define GATES 96            // 3*OUTF
#define NKC 8               // K chunks of 4 (K=32)
#define NJT 6               // 16-col tiles over 96 gate cols
#define FRAGS (NKC * NJT * 32)   // 1536 v2f per weight matrix

__device__ __forceinline__ float sigmoidf_(float x) {
    return 1.0f / (1.0f + __expf(-x));
}

// ------------------------- zero / copy helpers -----------------------------
__global__ void ggc_zero_kernel(float* __restrict__ p, int n) {
    int i = blockIdx.x * blockDim.x + threadIdx.x;
    for (; i < n; i += gridDim.x * blockDim.x) p[i] = 0.0f;
}

__global__ void ggc_copy_kernel(float* __restrict__ dstp,
                                const float* __restrict__ srcp, int n) {
    int i = blockIdx.x * blockDim.x + threadIdx.x;
    for (; i < n; i += gridDim.x * blockDim.x) dstp[i] = srcp[i];
}

// ------------------------- edge message + scatter --------------------------
// Grid-stride wave-per-edge. Lane j computes m[e][j] = sum_i h[s][i]*W[i][j]
// and atomically adds into a[dst][j]. Weight tables (4 x 32 x 32 f32 = 16KB)
// staged in LDS once per block (amortized over ~E/totalWaves edges per wave).
__global__ __launch_bounds__(256)
void ggc_edge_kernel(const float* __restrict__ h,
                     const int* __restrict__ src,
                     const int* __restrict__ dst,
                     const int* __restrict__ etypes,
                     const float* __restrict__ wtab,  // (4, 32*32)
                     float* __restrict__ a,
                     int nEdges) {
    __shared__ float Ws[4 * OUTF * OUTF];
    for (int i = threadIdx.x; i < 4 * OUTF * OUTF; i += blockDim.x)
        Ws[i] = wtab[i];
    __syncthreads();

    const int lane = threadIdx.x & 31;
    const int wavesPerBlock = blockDim.x >> 5;
    const int totalWaves = gridDim.x * wavesPerBlock;
    int e = blockIdx.x * wavesPerBlock + (threadIdx.x >> 5);

    for (; e < nEdges; e += totalWaves) {
        const int s = src[e];
        const int d = dst[e];
        const int t = etypes[e];

        const float hv = h[(size_t)s * OUTF + lane];
        const float* __restrict__ W = Ws + t * (OUTF * OUTF);

        float acc = 0.0f;
#pragma unroll
        for (int i = 0; i < OUTF; ++i) {
            const float hs = __shfl(hv, i, 32);       // broadcast h[s][i]
            acc = fmaf(hs, W[i * OUTF + lane], acc);  // W[i][lane]
        }
        atomicAdd(&a[(size_t)d * OUTF + lane], acc);  // global_atomic_add_f32
    }
}

// ------------------------- GRU cell via f32 WMMA ---------------------------
// gi = a @ w_ih^T + b_ih ; gh = h @ w_hh^T + b_hh   (N x 96 each)
// One wave per 16-node tile. D = A(16x4) x B(4x16) + C chained over K=32.
// A-frag (32-bit A 16x4): lanes 0-15 M=lane K={0,1}; lanes 16-31 K={2,3}.
// B-frag (4x16): lane holds col=lane&15, rows k0=(kc*4 + (lane>>4)*2)+{0,1}.
// LDS holds B-fragments pre-swizzled: bfrag[kc][jt][lane] = {w[g][k0],w[g][k0+1]}
// so each B operand is a single aligned ds_load_b64, conflict-free.
__global__ __launch_bounds__(256)
void ggc_gru_kernel(const float* __restrict__ a,
                    const float* __restrict__ hin,
                    const float* __restrict__ w_ih,   // (96,32) row-major
                    const float* __restrict__ w_hh,   // (96,32)
                    const float* __restrict__ b_ih,   // (96)
                    const float* __restrict__ b_hh,   // (96)
                    float* __restrict__ hout,
                    int nNodes, int nTiles) {
    __shared__ v2f bfragIH[FRAGS];    // 12 KB
    __shared__ v2f bfragHH[FRAGS];    // 12 KB
    __shared__ float bihS[GATES];
    __shared__ float bhhS[GATES];

    // Stage weights in fragment-major order. Source reads are contiguous 8B.
    for (int idx = threadIdx.x; idx < FRAGS; idx += blockDim.x) {
        const int ln = idx & 31;
        const int jt = (idx >> 5) % NJT;
        const int kc = idx / (NJT * 32);
        const int k0 = kc * 4 + ((ln >> 4) << 1);
        const int g  = jt * 16 + (ln & 15);
        bfragIH[idx] = *(const v2f*)(w_ih + g * OUTF + k0);
        bfragHH[idx] = *(const v2f*)(w_hh + g * OUTF + k0);
    }
    for (int idx = threadIdx.x; idx < GATES; idx += blockDim.x) {
        bihS[idx] = b_ih[idx];
        bhhS[idx] = b_hh[idx];
    }
    __syncthreads();

    const int tile = blockIdx.x * (blockDim.x >> 5) + (threadIdx.x >> 5);
    if (tile >= nTiles) return;        // wave-uniform exit: EXEC stays full

    const int lane = threadIdx.x & 31;
    const int r15  = lane & 15;
    const int koff = (lane >> 4) << 1;             // 0 or 2

    int rowA = tile * 16 + r15;
    if (rowA >= nNodes) rowA = nNodes - 1;         // clamp, no divergence

    // Preload A fragments for both GEMM inputs (a and h), K=32 in 8 chunks.
    v2f aF[NKC], hF[NKC];
#pragma unroll
    for (int kc = 0; kc < NKC; ++kc) {
        const size_t base = (size_t)rowA * OUTF + kc * 4 + koff;
        aF[kc] = *(const v2f*)(a + base);
        hF[kc] = *(const v2f*)(hin + base);
    }

#pragma unroll
    for (int jc = 0; jc < 2; ++jc) {               // output col halves
        v8f gi[3], gh[3];
#pragma unroll
        for (int t = 0; t < 3; ++t) {              // t=0:r  t=1:z  t=2:n gate
            const int jt = jc + 2 * t;
            const float bi = bihS[jt * 16 + r15];
            const float bh = bhhS[jt * 16 + r15];
#pragma unroll
            for (int v = 0; v < 8; ++v) { gi[t][v] = bi; gh[t][v] = bh; }
        }

#pragma unroll
        for (int kc = 0; kc < NKC; ++kc) {
#pragma unroll
            for (int t = 0; t < 3; ++t) {
                const int jt = jc + 2 * t;
                const int fidx = (kc * NJT + jt) * 32 + lane;
                const v2f bwi = bfragIH[fidx];     // one ds_load_b64
                gi[t] = __builtin_amdgcn_wmma_f32_16x16x4_f32(
                    false, aF[kc], false, bwi, (short)0, gi[t], false, false);
                const v2f bwh = bfragHH[fidx];     // one ds_load_b64
                gh[t] = __builtin_amdgcn_wmma_f32_16x16x4_f32(
                    false, hF[kc], false, bwh, (short)0, gh[t], false, false);
            }
        }

        // Fused GRU elementwise on accumulator tiles (layout-aligned).
        const int col   = jc * 16 + r15;
        const int mBase = tile * 16 + ((lane >> 4) << 3);
#pragma unroll
        for (int v = 0; v < 8; ++v) {
            const int m = mBase + v;
            if (m < nNodes) {
                const float r = sigmoidf_(gi[0][v] + gh[0][v]);
                const float z = sigmoidf_(gi[1][v] + gh[1][v]);
                const float n = tanhf(gi[2][v] + r * gh[2][v]);
                const float hv = hin[(size_t)m * OUTF + col];
                hout[(size_t)m * OUTF + col] = (1.0f - z) * n + z * hv;
            }
        }
    }
}

// ---------------------------------------------------------------------------
extern "C" void kernel_launch(void* const* d_in, const int* in_sizes, int n_in,
                              void* d_out, int out_size, void* d_ws, size_t ws_size,
                              hipStream_t stream) {
    const float* feat   = (const float*)d_in[0];
    const int*   etypes = (const int*)d_in[1];
    const int*   src    = (const int*)d_in[2];
    const int*   dst    = (const int*)d_in[3];
    const float* wtab   = (const float*)d_in[4];
    const float* w_ih   = (const float*)d_in[5];
    const float* w_hh   = (const float*)d_in[6];
    const float* b_ih   = (const float*)d_in[7];
    const float* b_hh   = (const float*)d_in[8];

    const int N = in_sizes[0] / OUTF;
    const int E = in_sizes[1];

    float* h = (float*)d_ws;                 // N*32 floats
    float* a = h + (size_t)N * OUTF;         // N*32 floats
    float* out = (float*)d_out;

    const int nElems = N * OUTF;
    const int cpBlocks = (nElems + 255) / 256;
    ggc_copy_kernel<<<cpBlocks, 256, 0, stream>>>(h, feat, nElems);

    const int nTiles    = (N + 15) / 16;
    const int gruBlocks = (nTiles + 7) / 8;  // 8 waves / 256-thread block

    // Grid-stride edge kernel: amortize 16KB LDS staging over ~E/32768 edges
    // per wave while keeping enough waves to saturate the atomic/HBM path.
    int edgeBlocks = 4096;
    const int maxEdgeBlocks = (int)(((long long)E + 7) / 8);
    if (edgeBlocks > maxEdgeBlocks) edgeBlocks = maxEdgeBlocks;
    if (edgeBlocks < 1) edgeBlocks = 1;

    for (int step = 0; step < 5; ++step) {
        ggc_zero_kernel<<<cpBlocks, 256, 0, stream>>>(a, nElems);
        ggc_edge_kernel<<<edgeBlocks, 256, 0, stream>>>(h, src, dst, etypes,
                                                        wtab, a, E);
        float* hdst = (step == 4) ? out : h;
        ggc_gru_kernel<<<gruBlocks, 256, 0, stream>>>(a, h, w_ih, w_hh,
                                                      b_ih, b_hh, hdst,
                                                      N, nTiles);
    }
}